// DeepgazeSpadeV2_39651138077208
// MI455X (gfx1250) — compile-verified
//
#include <hip/hip_runtime.h>

typedef __attribute__((ext_vector_type(2))) float v2f;
typedef __attribute__((ext_vector_type(8))) float v8f;

#define B     4
#define C     256
#define HF    48
#define WF    64
#define KTOT  (HF * WF)      // 3072 feature pixels
#define HIMG  384
#define WIMG  512
#define S     128
#define KCH   256            // k-chunk staged in LDS per iteration
#define LDSF_STRIDE (KCH + 2)
#define ROWS_PB 4            // image rows painted per block

// ---------------------------------------------------------------------------
// Phase 1: segment-sum as one-hot GEMM on WMMA.
//   sums[b][s][c] = sum_k (ids[k]==s) * feats[b][c][k]
// Grid: B * (C/16) blocks, 256 threads = 8 waves; wave w owns S-tile [16w,16w+16).
// ---------------------------------------------------------------------------
__global__ __launch_bounds__(256) void seg_reduce_wmma(
    const float* __restrict__ feats, const long long* __restrict__ seg,
    float* __restrict__ sums)
{
    __shared__ int   ids[KTOT];                 // 12 KB
    __shared__ float ldsF[16 * LDSF_STRIDE];    // ~16.1 KB, padded stride

    const int tid  = threadIdx.x;
    const int b    = blockIdx.x >> 4;           // C/16 == 16 c-tiles per batch
    const int c0   = (blockIdx.x & 15) << 4;
    const int lane = tid & 31;
    const int wave = tid >> 5;
    const int s0   = wave << 4;

    // nearest-downsample segmap -> 48x64 ids (src = seg[h*8, w*8]), clipped
    const long long* segb = seg + (long long)b * HIMG * WIMG;
    for (int k = tid; k < KTOT; k += 256) {
        int h = k >> 6, w = k & 63;
        int v = (int)segb[(h * 8) * WIMG + (w * 8)];
        v = v < 0 ? 0 : (v > S - 1 ? S - 1 : v);
        ids[k] = v;
    }

    const float* fb   = feats + ((long long)b * C + c0) * KTOT;
    const int    m    = s0 + (lane & 15);       // segment row this lane matches
    const int    koff = (lane >> 4) << 1;       // 0 (lanes 0-15) or 2 (16-31)
    v8f acc = {0.f, 0.f, 0.f, 0.f, 0.f, 0.f, 0.f, 0.f};

    for (int k0 = 0; k0 < KTOT; k0 += KCH) {
        __syncthreads();                        // WAR guard on ldsF
        for (int i = tid; i < 16 * KCH; i += 256) {
            int cl = i >> 8, kk = i & (KCH - 1);
            ldsF[cl * LDSF_STRIDE + kk] = fb[(long long)cl * KTOT + k0 + kk];
        }
        __syncthreads();

        const float* frow = &ldsF[(lane & 15) * LDSF_STRIDE];
        for (int kk = 0; kk < KCH; kk += 4) {
            // A (16x4 one-hot) per ISA layout: lane holds K=koff, koff+1
            int kb = k0 + kk + koff;
            v2f a, bv;
            a.x = (ids[kb]     == m) ? 1.0f : 0.0f;
            a.y = (ids[kb + 1] == m) ? 1.0f : 0.0f;
            // B (4x16): VGPR0 -> K=koff row, VGPR1 -> K=koff+1 row, N = lane&15
            bv.x = frow[kk + koff];
            bv.y = frow[kk + koff + 1];
            acc = __builtin_amdgcn_wmma_f32_16x16x4_f32(
                      false, a, false, bv, (short)0, acc, false, false);
        }
    }

    // D tile (16x16 f32): VGPR j -> M = j (lanes 0-15) / j+8 (lanes 16-31)
    float* sb  = sums + (long long)b * S * C;
    const int cc   = c0 + (lane & 15);
    const int srow = s0 + ((lane >> 4) << 3);
#pragma unroll
    for (int j = 0; j < 8; ++j)
        sb[(srow + j) * C + cc] = acc[j];
}

// ---------------------------------------------------------------------------
// Phase 2: counts + normalize; write avg TRANSPOSED as avgT[b][c][s] so the
// paint phase's LDS gather banks on s (conflict-friendly). One block per batch.
// ---------------------------------------------------------------------------
__global__ __launch_bounds__(256) void seg_normalize(
    const long long* __restrict__ seg, const float* __restrict__ sums,
    float* __restrict__ avgT)
{
    __shared__ int cnt[S];
    const int tid = threadIdx.x;
    const int b   = blockIdx.x;

    for (int i = tid; i < S; i += 256) cnt[i] = 0;
    __syncthreads();

    const long long* segb = seg + (long long)b * HIMG * WIMG;
    for (int k = tid; k < KTOT; k += 256) {
        int h = k >> 6, w = k & 63;
        int v = (int)segb[(h * 8) * WIMG + (w * 8)];
        v = v < 0 ? 0 : (v > S - 1 ? S - 1 : v);
        atomicAdd(&cnt[v], 1);
    }
    __syncthreads();

    const float* sb = sums + (long long)b * S * C;
    float*       ab = avgT + (long long)b * S * C;
    for (int i = tid; i < S * C; i += 256) {
        int c = i >> 7, s = i & (S - 1);
        int n = cnt[s];
        ab[i] = (n > 0) ? sb[s * C + c] / (float)n : 0.0f;   // empty seg -> 0
    }
}

// ---------------------------------------------------------------------------
// Phase 3: repaint (the 805 MB stream). Block = (batch, 4 image rows).
// avgT[b] (128 KB) lives in dynamic LDS; output via non-temporal stores.
// ---------------------------------------------------------------------------
__global__ __launch_bounds__(256) void seg_paint(
    const long long* __restrict__ seg, const float* __restrict__ avgT,
    float* __restrict__ out)
{
    extern __shared__ float lds[];              // S*C floats + ROWS_PB*WIMG ints
    int* lseg = (int*)(lds + S * C);

    const int tid = threadIdx.x;
    const int b   = blockIdx.x / (HIMG / ROWS_PB);
    const int rg  = blockIdx.x % (HIMG / ROWS_PB);
    const int h0  = rg * ROWS_PB;

    const float* ab = avgT + (long long)b * S * C;
    for (int i = tid; i < S * C; i += 256) lds[i] = ab[i];

    const long long* segb = seg + (long long)b * HIMG * WIMG;
    for (int i = tid; i < ROWS_PB * WIMG; i += 256) {
        int h = h0 + i / WIMG, w = i % WIMG;
        int v = (int)segb[h * WIMG + w];
        v = v < 0 ? 0 : (v > S - 1 ? S - 1 : v);
        lseg[i] = v;
    }
    __syncthreads();

    float* ob = out + (long long)b * C * HIMG * WIMG;
    for (int c = 0; c < C; ++c) {
        const float* arow = &lds[c * S];        // bank index = s & 63
        for (int r = 0; r < ROWS_PB; ++r) {
            long long rowbase = (long long)c * HIMG * WIMG
                              + (long long)(h0 + r) * WIMG;
            int sb = r * WIMG;
#pragma unroll
            for (int half = 0; half < 2; ++half) {
                int w = tid + half * 256;       // coalesced 128B/wave stores
                __builtin_nontemporal_store(arow[lseg[sb + w]],
                                            &ob[rowbase + w]);
            }
        }
    }
}

// ---------------------------------------------------------------------------
extern "C" void kernel_launch(void* const* d_in, const int* in_sizes, int n_in,
                              void* d_out, int out_size, void* d_ws, size_t ws_size,
                              hipStream_t stream)
{
    const float*     feats = (const float*)d_in[0];      // [B,C,48,64] f32
    const long long* seg   = (const long long*)d_in[1];  // [B,384,512] i64
    float*           out   = (float*)d_out;              // [B,C,384,512] f32

    float* sums = (float*)d_ws;                          // [B,S,C]  512 KB
    float* avgT = sums + (size_t)B * S * C;              // [B,C,S]  512 KB

    seg_reduce_wmma<<<B * (C / 16), 256, 0, stream>>>(feats, seg, sums);
    seg_normalize  <<<B,            256, 0, stream>>>(seg, sums, avgT);

    size_t shmem = (size_t)(S * C + ROWS_PB * WIMG) * sizeof(float); // ~136 KB
    seg_paint<<<B * (HIMG / ROWS_PB), 256, shmem, stream>>>(seg, avgT, out);
}